// BKT_84490596646991
// MI455X (gfx1250) — compile-verified
//
#include <hip/hip_runtime.h>
#include <stdint.h>

// ---------------------------------------------------------------------------
// BKT forward scan for MI455X (gfx1250) — LDS-resident state version.
//
// Critical path = 512 strictly-dependent per-student state RMWs. We exploit
// CDNA5's 320KB-per-WGP LDS to keep EVERYTHING on-chip per block of 32
// students (one wave32):
//     state  : 32 students x 2048 skills x f32          = 256 KB
//     params : k0/t/g/s, 4 x 2048 x f32                 =  32 KB
//     tiles  : double-buffered skill/response chunks    =  18 KB
//                                                  total = 313 KB  (< 320 KB)
// All global loads move through CDNA5 async global->LDS DMA
// (ASYNCcnt / s_wait_asynccnt), double-buffered so the next chunk streams in
// while the current chunk's serial recurrence runs (only 1 wave/SIMD here,
// so overlap must come from async DMA, not co-resident waves).
// ---------------------------------------------------------------------------

#define B_STUDENTS 4096
#define T_STEPS    512
#define K_SKILLS   2048
#define BLOCK      32                 // one wave32; 32 students per block
#define CT         32                 // timesteps per staged chunk
#define TSTRIDE    36                 // dwords/row: 144B => 16B-aligned rows for b128 DMA
#define NCHUNK     (T_STEPS / CT)     // 16

#if defined(__gfx1250__) && __has_builtin(__builtin_amdgcn_global_load_async_to_lds_b32)
#define ASYNC_LDS 1
#else
#define ASYNC_LDS 0
#endif

#if ASYNC_LDS && __has_builtin(__builtin_amdgcn_global_load_async_to_lds_b128)
#define HAVE_B128 1
#define PIPE 16                        // async load-ops per chunk (in-order completion)
#else
#define HAVE_B128 0
#define PIPE 0                         // fallback: fully drain each chunk
#endif

typedef int v4i __attribute__((ext_vector_type(4)));
typedef __attribute__((address_space(1))) int gint_t;   // global int
typedef __attribute__((address_space(3))) int lint_t;   // LDS int
typedef __attribute__((address_space(1))) v4i gv4i_t;   // global int4
typedef __attribute__((address_space(3))) v4i lv4i_t;   // LDS int4

// ---- async DMA helpers -----------------------------------------------------
__device__ __forceinline__ void async_g2l_b128(const void* g, void* l) {
#if HAVE_B128
    __builtin_amdgcn_global_load_async_to_lds_b128((gv4i_t*)g, (lv4i_t*)l, 0, 0);
#elif ASYNC_LDS
    __builtin_amdgcn_global_load_async_to_lds_b32((gint_t*)g + 0, (lint_t*)l + 0, 0, 0);
    __builtin_amdgcn_global_load_async_to_lds_b32((gint_t*)g + 1, (lint_t*)l + 1, 0, 0);
    __builtin_amdgcn_global_load_async_to_lds_b32((gint_t*)g + 2, (lint_t*)l + 2, 0, 0);
    __builtin_amdgcn_global_load_async_to_lds_b32((gint_t*)g + 3, (lint_t*)l + 3, 0, 0);
#else
    *(float4*)l = *(const float4*)g;
#endif
}

template <int N>
__device__ __forceinline__ void wait_async() {
#if ASYNC_LDS
#if __has_builtin(__builtin_amdgcn_s_wait_asynccnt)
    __builtin_amdgcn_s_wait_asynccnt(N);
#else
    asm volatile("s_wait_asynccnt %0" :: "i"(N) : "memory");
#endif
#endif
}

__device__ __forceinline__ void wait_ds0() {
#if __has_builtin(__builtin_amdgcn_s_wait_dscnt)
    __builtin_amdgcn_s_wait_dscnt(0);
#else
    asm volatile("s_wait_dscnt 0" ::: "memory");
#endif
}

// Stage one [BLOCK x CT] chunk of skills+responses into an LDS tile buffer.
// Wave issues 16 b128 DMA ops (8 per array); rows are 16B-aligned (TSTRIDE=36).
__device__ __forceinline__ void stage_tile(const int* skills, const float* responses,
                                           int b0, int t0, int tid,
                                           int* skt, float* rst) {
#pragma unroll
    for (int i = 0; i < 8; ++i) {
        int fi  = i * BLOCK + tid;     // 0..255 float4s
        int row = fi >> 3;             // student row within block
        int c4  = fi & 7;              // float4 column
        size_t gofs = (size_t)(b0 + row) * T_STEPS + (size_t)t0 + (size_t)(c4 << 2);
        async_g2l_b128(skills    + gofs, skt + row * TSTRIDE + (c4 << 2));
        async_g2l_b128(responses + gofs, rst + row * TSTRIDE + (c4 << 2));
    }
}

// ---- main kernel -----------------------------------------------------------
__global__ void bkt_main(const int*   __restrict__ skills,     // [B,T]
                         const float* __restrict__ responses,  // [B,T]
                         const float* __restrict__ k0g,
                         const float* __restrict__ tg,
                         const float* __restrict__ gg_,
                         const float* __restrict__ sg,
                         float* __restrict__ out) {             // [B,T]
    extern __shared__ float smem[];
    float* stateL = smem;                                   // BLOCK*K  (256 KB)
    float* k0s    = stateL + BLOCK * K_SKILLS;              // K
    float* ts     = k0s + K_SKILLS;
    float* gs     = ts  + K_SKILLS;
    float* ss     = gs  + K_SKILLS;
    int*   skt0   = (int*)(ss + K_SKILLS);                  // tile buffers
    float* rst0   = (float*)(skt0 + BLOCK * TSTRIDE);
    int*   skt1   = (int*)(rst0 + BLOCK * TSTRIDE);
    float* rst1   = (float*)(skt1 + BLOCK * TSTRIDE);

    const int tid = threadIdx.x;
    const int b0  = blockIdx.x * BLOCK;

    // Kick off all startup DMA first so it overlaps the LDS state init below.
    // Params: 4 tables x 512 float4s, 16 f4 per lane per table.
#pragma unroll
    for (int i = 0; i < 16; ++i) {
        int fi = i * BLOCK + tid;                   // f4 index 0..511
        async_g2l_b128((const float4*)k0g + fi, (float4*)k0s + fi);
        async_g2l_b128((const float4*)tg  + fi, (float4*)ts  + fi);
        async_g2l_b128((const float4*)gg_ + fi, (float4*)gs  + fi);
        async_g2l_b128((const float4*)sg  + fi, (float4*)ss  + fi);
    }
    stage_tile(skills, responses, b0, 0,      tid, skt0, rst0);   // chunk 0
    stage_tile(skills, responses, b0, CT,     tid, skt1, rst1);   // chunk 1

    // Sentinel-init this wave's 256KB state slab while DMA streams in.
    // Rotate the f4 index per lane so lanes hit different bank groups
    // (unrotated, all 32 lanes of a wave collide on the same 4 banks).
    {
        float4* st4 = (float4*)stateL;
        const float4 sent = make_float4(-1.0f, -1.0f, -1.0f, -1.0f);
#pragma unroll 4
        for (int jj = 0; jj < K_SKILLS / 4; ++jj) {
            int j = (jj + tid) & (K_SKILLS / 4 - 1);
            st4[tid * (K_SKILLS / 4) + j] = sent;
        }
    }

    wait_async<0>();          // params + first two chunks resident
    __syncthreads();          // (single-wave workgroup: effectively free)

    float* myState = stateL + tid * K_SKILLS;

    for (int c = 0; c < NCHUNK; ++c) {
        // Ensure buffer (c&1) is resident. With b128 staging, async loads
        // complete in order, so <=PIPE outstanding == chunk c landed while
        // chunk c+1 may still be in flight.
        if (c + 1 < NCHUNK) wait_async<PIPE>(); else wait_async<0>();

        int*   skt = (c & 1) ? skt1 : skt0;
        float* rst = (c & 1) ? rst1 : rst0;
        const int t0 = c * CT;

        // Serial BKT recurrence, all operands in LDS.
#pragma unroll 4
        for (int tt = 0; tt < CT; ++tt) {
            int   sk  = skt[tid * TSTRIDE + tt];
            float res = rst[tid * TSTRIDE + tt];
            float lm  = myState[sk];                  // LDS-resident RMW
            float pL  = (lm >= 0.0f) ? lm : k0s[sk];  // first-seen -> prior k0
            float sv  = ss[sk];
            float gv  = gs[sk];
            float tv  = ts[sk];
            bool  corr = (res == 1.0f);
            float num = corr ? pL * (1.0f - sv)
                             : pL * sv;
            float den = corr ? (pL * (1.0f - sv) + (1.0f - pL) * gv)
                             : (pL * sv          + (1.0f - pL) * (1.0f - gv));
            float p   = num / den;
            myState[sk] = p + (1.0f - p) * tv;        // learning transition
            rst[tid * TSTRIDE + tt] = pL;             // stash for writeback
        }

        // Coalesced writeback: lane = timestep. LDS reads are conflict-free
        // (consecutive lanes -> consecutive banks); stores are 128B/wave,
        // nontemporal (streamed once, keep L2 for the input stream).
#pragma unroll 8
        for (int i = 0; i < BLOCK; ++i) {
            float v = rst[i * TSTRIDE + tid];
            __builtin_nontemporal_store(v, &out[(size_t)(b0 + i) * T_STEPS + (size_t)(t0 + tid)]);
        }

        // Drain our ds_loads of this buffer before DMA overwrites it.
        wait_ds0();
        if (c + 2 < NCHUNK) {
            stage_tile(skills, responses, b0, (c + 2) * CT, tid,
                       (c & 1) ? skt1 : skt0, (c & 1) ? rst1 : rst0);
        }
    }
}

// ---------------------------------------------------------------------------
extern "C" void kernel_launch(void* const* d_in, const int* in_sizes, int n_in,
                              void* d_out, int out_size, void* d_ws, size_t ws_size,
                              hipStream_t stream) {
    const int*   skills    = (const int*)  d_in[0];   // [B,T] int32
    const float* responses = (const float*)d_in[1];   // [B,T] f32
    const float* k0        = (const float*)d_in[2];   // [K]
    const float* t         = (const float*)d_in[3];   // [K]
    const float* g         = (const float*)d_in[4];   // [K]
    const float* s         = (const float*)d_in[5];   // [K]
    float*       out       = (float*)d_out;           // [B,T]

    // 256KB state + 32KB params + 18KB double-buffered tiles = 313344 B
    const size_t ldsBytes = (size_t)(BLOCK * K_SKILLS + 4 * K_SKILLS
                                     + 4 * BLOCK * TSTRIDE) * sizeof(float);

    // Permit >64KB dynamic LDS (no-op / ignored error if not required).
    (void)hipFuncSetAttribute((const void*)bkt_main,
                              hipFuncAttributeMaxDynamicSharedMemorySize,
                              (int)ldsBytes);

    hipLaunchKernelGGL(bkt_main, dim3(B_STUDENTS / BLOCK), dim3(BLOCK), ldsBytes, stream,
                       skills, responses, k0, t, g, s, out);
}